// DGCNN_62414464745554
// MI455X (gfx1250) — compile-verified
//
#include <hip/hip_runtime.h>
#include <math.h>

typedef float v2f __attribute__((ext_vector_type(2)));
typedef float v8f __attribute__((ext_vector_type(8)));

#define LEAKY_SLOPE 0.2f
#define BN_EPS 1e-5f

// ---------------------------------------------------------------------------
// Build stage-1 X matrix directly in padded layout.
// X0[d,p] (d<1024, p<342, ld=352) = tiled_flat[d*342+p], tiled[c,n]=x[b,c%2,n]
// ---------------------------------------------------------------------------
__global__ void build_x0_kernel(const float* __restrict__ x, float* __restrict__ X0,
                                int Bn)
{
    long i = (long)blockIdx.x * blockDim.x + threadIdx.x;
    long total = (long)Bn * 1024 * 342;
    if (i >= total) return;
    int p = (int)(i % 342);
    int d = (int)((i / 342) % 1024);
    int b = (int)(i / (342L * 1024));
    long f = (long)d * 342 + p;
    int c = (int)(f / 1024);
    int n = (int)(f % 1024);
    X0[(long)b * 1024 * 352 + (long)d * 352 + p] =
        x[(long)b * 2048 + (long)(c & 1) * 1024 + n];
}

__global__ void zero_kernel(float* __restrict__ p, long n)
{
    long i = (long)blockIdx.x * blockDim.x + threadIdx.x;
    if (i < n) p[i] = 0.f;
}

// Copy W[o, colOff + c] (o<O, c<C) into zero-padded Wp (Op x Cp row-major).
__global__ void pad_w_kernel(const float* __restrict__ W, float* __restrict__ Wp,
                             int O, int C, int ldw, int colOff, int Op, int Cp)
{
    long i = (long)blockIdx.x * blockDim.x + threadIdx.x;
    if (i >= (long)Op * Cp) return;
    int c = (int)(i % Cp);
    int o = (int)(i / Cp);
    Wp[i] = (o < O && c < C) ? W[(long)o * ldw + colOff + c] : 0.f;
}

// ---------------------------------------------------------------------------
// f32 WMMA GEMM, one wave32 per 32x32 C tile (2x2 register tiling).
// M, N multiples of 32; Kd multiple of 16.  No bounds checks anywhere:
// unconditional loads, EXEC untouched.  K unrolled x16 -> 16 chained
// V_WMMA_F32_16X16X4_F32 per iteration; each A/B fragment feeds 2 WMMAs
// (2x the compute intensity of 1x1 tiling).  Incremental pointers avoid
// per-iteration 64-bit multiplies.
//   ATRANS=false: C[m,n] = sum_k A[m,k]*B[k,n]   (A row-major MxK)
//   ATRANS=true : C[m,n] = sum_k A[k,m]*B[k,n]   (A row-major KxM)
// Fragment layouts (wave32):
//   A: lane l, vgpr j -> A[M=l%16, K=kb+(l/16)*2+j]
//   B: lane l, vgpr j -> B[K=kb+(l/16)*2+j, N=l%16]
//   C: lane l, vgpr r -> C[M=r+(l/16)*8,   N=l%16]
// ---------------------------------------------------------------------------
template <bool ATRANS>
__global__ void gemm_wmma_f32(const float* __restrict__ A, const float* __restrict__ B,
                              float* __restrict__ C, int batch,
                              int M, int N, int Kd,
                              int lda, int ldb, int ldc,
                              long sA, long sB, long sC)
{
    int wave = (int)(((long)blockIdx.x * blockDim.x + threadIdx.x) >> 5);
    int lane = threadIdx.x & 31;
    int tN = N >> 5;
    int tM = M >> 5;
    int per = tM * tN;
    int b = wave / per;
    if (b >= batch) return;                  // wave-uniform exit only
    int t = wave - b * per;
    int tm = (t / tN) << 5;
    int tn = (t % tN) << 5;
    const float* Ab = A + (long)b * sA;
    const float* Bb = B + (long)b * sB;
    float* Cb = C + (long)b * sC;
    int half = lane >> 4;
    int l16 = lane & 15;

    const float* a0p;
    const float* a1p;
    long aStep;
    if (ATRANS) {
        a0p = Ab + (long)(half * 2) * lda + tm + l16;   // advance over K rows
        a1p = a0p + 16;
        aStep = 4L * lda;
    } else {
        a0p = Ab + (long)(tm + l16) * lda + half * 2;   // advance over K cols
        a1p = a0p + 16L * lda;
        aStep = 4;
    }
    const float* b0p = Bb + (long)(half * 2) * ldb + tn + l16;
    const float* b1p = b0p + 16;
    long bStep = 4L * ldb;

    v8f acc00 = {}, acc01 = {}, acc10 = {}, acc11 = {};
    for (int k0 = 0; k0 < Kd; k0 += 16) {
        if (k0 + 16 < Kd) {                  // global_prefetch_b8 of upcoming B rows
            __builtin_prefetch(Bb + (long)(k0 + 16) * ldb + tn, 0, 1);
        }
#pragma unroll
        for (int s = 0; s < 4; ++s) {
            v2f a0, a1, b0, b1;
            if (ATRANS) {
                a0[0] = a0p[0]; a0[1] = a0p[lda];
                a1[0] = a1p[0]; a1[1] = a1p[lda];
            } else {
                a0 = *(const v2f*)a0p;       // contiguous 64-bit loads
                a1 = *(const v2f*)a1p;
            }
            b0[0] = b0p[0]; b0[1] = b0p[ldb];
            b1[0] = b1p[0]; b1[1] = b1p[ldb];
            a0p += aStep; a1p += aStep; b0p += bStep; b1p += bStep;
            acc00 = __builtin_amdgcn_wmma_f32_16x16x4_f32(false, a0, false, b0,
                                                          (short)0, acc00, false, false);
            acc01 = __builtin_amdgcn_wmma_f32_16x16x4_f32(false, a0, false, b1,
                                                          (short)0, acc01, false, false);
            acc10 = __builtin_amdgcn_wmma_f32_16x16x4_f32(false, a1, false, b0,
                                                          (short)0, acc10, false, false);
            acc11 = __builtin_amdgcn_wmma_f32_16x16x4_f32(false, a1, false, b1,
                                                          (short)0, acc11, false, false);
        }
    }
#pragma unroll
    for (int r = 0; r < 8; ++r) {
        int m0 = tm + r + half * 8;
        Cb[(long)m0 * ldc + tn + l16]             = acc00[r];
        Cb[(long)m0 * ldc + tn + 16 + l16]        = acc01[r];
        Cb[(long)(m0 + 16) * ldc + tn + l16]      = acc10[r];
        Cb[(long)(m0 + 16) * ldc + tn + 16 + l16] = acc11[r];
    }
}

// ---------------------------------------------------------------------------
// Top-k (k=20) neighbors from Gram matrix (padded ld).
// pd[p,q] = 2*G[p,q] - G[p,p] - G[q,q]; tie-break = lower index (lax.top_k).
// One wave32 per (b,p) row, shuffle butterfly reduction.
// ---------------------------------------------------------------------------
__global__ void topk_kernel(const float* __restrict__ G, int* __restrict__ idx,
                            int Bn, int P, int ldG, int k)
{
    int wave = (int)(((long)blockIdx.x * blockDim.x + threadIdx.x) >> 5);
    int lane = threadIdx.x & 31;
    if (wave >= Bn * P) return;
    int b = wave / P, p = wave - b * P;
    const float* Gb = G + (long)b * ldG * ldG;
    const float* Gp = Gb + (long)p * ldG;
    float gpp = Gp[p];
    unsigned taken = 0u;
    for (int sel = 0; sel < k; ++sel) {
        float bvv = -3.4e38f;
        int bq = 0x7fffffff;
        int s = 0;
        for (int q = lane; q < P; q += 32, ++s) {
            if (taken & (1u << s)) continue;
            float v = 2.f * Gp[q] - gpp - Gb[(long)q * ldG + q];
            if (v > bvv || (v == bvv && q < bq)) { bvv = v; bq = q; }
        }
        for (int off = 16; off > 0; off >>= 1) {
            float ov = __shfl_xor(bvv, off, 32);
            int   oq = __shfl_xor(bq, off, 32);
            if (ov > bvv || (ov == bvv && oq < bq)) { bvv = ov; bq = oq; }
        }
        if (lane == 0) idx[((long)b * P + p) * k + sel] = bq;
        if ((bq & 31) == lane) taken |= 1u << (bq >> 5);
    }
}

// S = (hasCtr ? Zc : 0) - Zd   over the full padded extent
__global__ void s_kernel(const float* __restrict__ Zd, const float* __restrict__ Zc,
                         float* __restrict__ S, long n, int hasCtr)
{
    long i = (long)blockIdx.x * blockDim.x + threadIdx.x;
    if (i >= n) return;
    S[i] = (hasCtr ? Zc[i] : 0.f) - Zd[i];
}

// ---------------------------------------------------------------------------
// BN stats over y[b,o,p,k] = Zd[b,o,idx[b,p,k]] + S[b,o,p]   (one block per o)
// ---------------------------------------------------------------------------
__global__ void bn_stats_graph(const float* __restrict__ Zd, const float* __restrict__ S,
                               const int* __restrict__ idx,
                               float* __restrict__ mean, float* __restrict__ istd,
                               int Bn, int O, int P, int k, int ldZ, long sZb)
{
    int o = blockIdx.x;
    int tid = threadIdx.x;
    float sum = 0.f, sq = 0.f;
    int total = Bn * P * k;
    for (int i = tid; i < total; i += blockDim.x) {
        int kk = i % k;
        int p = (i / k) % P;
        int b = i / (k * P);
        int q = idx[((long)b * P + p) * k + kk];
        long base = (long)b * sZb + (long)o * ldZ;
        float y = Zd[base + q] + S[base + p];
        sum += y; sq += y * y;
    }
    __shared__ float sh1[256], sh2[256];
    sh1[tid] = sum; sh2[tid] = sq;
    __syncthreads();
    for (int st = 128; st > 0; st >>= 1) {
        if (tid < st) { sh1[tid] += sh1[tid + st]; sh2[tid] += sh2[tid + st]; }
        __syncthreads();
    }
    if (tid == 0) {
        float m = sh1[0] / (float)total;
        float v = sh2[0] / (float)total - m * m;   // biased variance (jnp.var)
        mean[o] = m;
        istd[o] = rsqrtf(v + BN_EPS);
    }
}

// Fused BN + leaky-ReLU + max over k; output written with generic strides so it
// lands directly in the NEXT stage's padded X layout (or x4's natural layout).
__global__ void apply_graph(const float* __restrict__ Zd, const float* __restrict__ S,
                            const int* __restrict__ idx,
                            const float* __restrict__ mean, const float* __restrict__ istd,
                            const float* __restrict__ gam, const float* __restrict__ bet,
                            float* __restrict__ out, int Bn, int O, int P, int k,
                            int ldZ, long sZb, long osB, int osO, int osP)
{
    long i = (long)blockIdx.x * blockDim.x + threadIdx.x;
    if (i >= (long)Bn * O * P) return;
    int p = (int)(i % P);
    int o = (int)((i / P) % O);
    int b = (int)(i / ((long)P * O));
    float m = mean[o], is = istd[o], gg = gam[o], bb = bet[o];
    long base = (long)b * sZb + (long)o * ldZ;
    float sv = S[base + p];
    const int* ip = idx + ((long)b * P + p) * k;
    float best = -3.4e38f;
    for (int kk = 0; kk < k; ++kk) {
        float y = Zd[base + ip[kk]] + sv;
        float z = (y - m) * is * gg + bb;
        z = z >= 0.f ? z : LEAKY_SLOPE * z;
        best = fmaxf(best, z);
    }
    out[(long)b * osB + (long)o * osO + (long)p * osP] = best;
}

// Head gather: z[b,c,l] (B,400,64) = concat(x3[:, :300, :64], x4[:, :100, :64])
// x3 lives in padded X3p (688 x 320): value(b, c, l) = X3p[b][row=l][col=c]
__global__ void headz_kernel(const float* __restrict__ X3p, const float* __restrict__ x4,
                             float* __restrict__ z, int Bn)
{
    long i = (long)blockIdx.x * blockDim.x + threadIdx.x;
    if (i >= (long)Bn * 400 * 64) return;
    int l = (int)(i % 64);
    int c = (int)((i / 64) % 400);
    int b = (int)(i / (64 * 400));
    float v;
    if (c < 300) v = X3p[(long)b * 688 * 320 + (long)l * 320 + c];
    else         v = x4[((long)b * 500 + (c - 300)) * 300 + l];
    z[i] = v;
}

__global__ void bn_stats_head(const float* __restrict__ Y, float* __restrict__ mean,
                              float* __restrict__ istd, int Bn, int O, int L, long sYb)
{
    int o = blockIdx.x;
    int tid = threadIdx.x;
    float sum = 0.f, sq = 0.f;
    int total = Bn * L;
    for (int i = tid; i < total; i += blockDim.x) {
        int b = i / L, l = i % L;
        float y = Y[(long)b * sYb + (long)o * L + l];
        sum += y; sq += y * y;
    }
    __shared__ float sh1[256], sh2[256];
    sh1[tid] = sum; sh2[tid] = sq;
    __syncthreads();
    for (int st = 128; st > 0; st >>= 1) {
        if (tid < st) { sh1[tid] += sh1[tid + st]; sh2[tid] += sh2[tid + st]; }
        __syncthreads();
    }
    if (tid == 0) {
        float m = sh1[0] / (float)total;
        float v = sh2[0] / (float)total - m * m;
        mean[o] = m;
        istd[o] = rsqrtf(v + BN_EPS);
    }
}

__global__ void head_out_kernel(const float* __restrict__ Y,
                                const float* __restrict__ mean, const float* __restrict__ istd,
                                const float* __restrict__ gam, const float* __restrict__ bet,
                                float* __restrict__ out, int Bn, int O, int L, long sYb)
{
    int i = blockIdx.x * blockDim.x + threadIdx.x;
    if (i >= Bn * O) return;
    int o = i % O, b = i / O;
    float m = mean[o], is = istd[o], gg = gam[o], bb = bet[o];
    float best = -3.4e38f, sum = 0.f;
    for (int l = 0; l < L; ++l) {
        float y = Y[(long)b * sYb + (long)o * L + l];
        float z = (y - m) * is * gg + bb;
        z = z >= 0.f ? z : LEAKY_SLOPE * z;
        best = fmaxf(best, z);
        sum += z;
    }
    out[(long)b * 2 * O + o] = best;
    out[(long)b * 2 * O + O + o] = sum / (float)L;
}

// ---------------------------------------------------------------------------
extern "C" void kernel_launch(void* const* d_in, const int* in_sizes, int n_in,
                              void* d_out, int out_size, void* d_ws, size_t ws_size,
                              hipStream_t stream)
{
    (void)in_sizes; (void)n_in; (void)out_size; (void)ws_size;
    const int Bn = 8, kNN = 20;
    const float* x0 = (const float*)d_in[0];
    const float* W[4]  = {(const float*)d_in[1],  (const float*)d_in[4],
                          (const float*)d_in[7],  (const float*)d_in[10]};
    const float* Gm[4] = {(const float*)d_in[2],  (const float*)d_in[5],
                          (const float*)d_in[8],  (const float*)d_in[11]};
    const float* Bt[4] = {(const float*)d_in[3],  (const float*)d_in[6],
                          (const float*)d_in[9],  (const float*)d_in[12]};
    const float* W5 = (const float*)d_in[13];
    const float* g5 = (const float*)d_in[14];
    const float* b5 = (const float*)d_in[15];

    // true / padded stage dims.  M,N padded to x32; K padded to x16.
    const int Pp[4]  = {342, 500, 684, 300};
    const int Oo[4]  = {500, 684, 300, 500};
    const int Cc[4]  = {500, 342, 300, 500};
    const int ldW[4] = {500, 684, 300, 500};
    const int D16[4] = {1024, 352, 512, 688};   // K (rows of X), x16
    const int P32[4] = {352, 512, 704, 320};    // N / gram M, x32
    const int O32[4] = {512, 704, 320, 512};    // Z rows, x32
    const int C16[4] = {512, 352, 304, 512};    // Z K, x16

    char* ws = (char*)d_ws;
    size_t off = 0;
    auto alloc = [&](size_t bytes) -> void* {
        void* p = ws + off;
        off += (bytes + 255) & ~(size_t)255;
        return p;
    };
    // padded X buffers (stage s reads Xp[s]: D16[s] x P32[s], ld = P32[s])
    float* X0p = (float*)alloc((size_t)Bn * 1024 * 352 * 4);
    float* X1p = (float*)alloc((size_t)Bn * 352 * 512 * 4);
    float* X2p = (float*)alloc((size_t)Bn * 512 * 704 * 4);
    float* X3p = (float*)alloc((size_t)Bn * 688 * 320 * 4);
    float* X4  = (float*)alloc((size_t)Bn * 500 * 300 * 4);   // natural (B,500,300)
    float* G   = (float*)alloc((size_t)Bn * 704 * 704 * 4);   // padded P32 x P32
    int*   idx = (int*)  alloc((size_t)Bn * 704 * kNN * 4);
    float* Zd  = (float*)alloc((size_t)Bn * 704 * 512 * 4);   // >= max O32*P32
    float* Zc  = (float*)alloc((size_t)Bn * 704 * 512 * 4);
    float* S   = (float*)alloc((size_t)Bn * 704 * 512 * 4);
    float* Wp  = (float*)alloc((size_t)704 * 512 * 4);        // >= max O32*C16
    float* Wcp = (float*)alloc((size_t)704 * 352 * 4);
    float* W5p = (float*)alloc((size_t)416 * 400 * 4);        // head W padded M->416
    float* Zh  = (float*)alloc((size_t)Bn * 400 * 64 * 4);
    float* Yh  = (float*)alloc((size_t)Bn * 416 * 64 * 4);
    float* mean = (float*)alloc(1024 * 4);
    float* istd = (float*)alloc(1024 * 4);

    // zero the padded X buffers (pad rows/cols must be 0 for the K-loops)
    {
        long n0 = (long)Bn * 1024 * 352, n1 = (long)Bn * 352 * 512;
        long n2 = (long)Bn * 512 * 704,  n3 = (long)Bn * 688 * 320;
        zero_kernel<<<(int)((n0 + 255) / 256), 256, 0, stream>>>(X0p, n0);
        zero_kernel<<<(int)((n1 + 255) / 256), 256, 0, stream>>>(X1p, n1);
        zero_kernel<<<(int)((n2 + 255) / 256), 256, 0, stream>>>(X2p, n2);
        zero_kernel<<<(int)((n3 + 255) / 256), 256, 0, stream>>>(X3p, n3);
    }
    {   // stage-1 X in padded layout
        long total = (long)Bn * 1024 * 342;
        build_x0_kernel<<<(int)((total + 255) / 256), 256, 0, stream>>>(x0, X0p, Bn);
    }

    const float* Xp[4] = {X0p, X1p, X2p, X3p};
    float* Xnext[4] = {X1p, X2p, X3p, X4};

    for (int s = 0; s < 4; ++s) {
        int P = Pp[s], O = Oo[s], C = Cc[s];
        int d16 = D16[s], p32 = P32[s], o32 = O32[s], c16 = C16[s];

        {   // Gram = X^T X  (p32 x p32, K = d16) -- guard-free 2x2 WMMA tiles
            long waves = (long)Bn * (p32 >> 5) * (p32 >> 5);
            gemm_wmma_f32<true><<<(int)((waves + 7) / 8), 256, 0, stream>>>(
                Xp[s], Xp[s], G, Bn, p32, p32, d16, p32, p32, p32,
                (long)d16 * p32, (long)d16 * p32, (long)p32 * p32);
        }
        {   // top-k neighbors
            long waves = (long)Bn * P;
            topk_kernel<<<(int)((waves + 7) / 8), 256, 0, stream>>>(G, idx, Bn, P, p32, kNN);
        }
        {   // pad weights, then Zd = Wp @ X (o32 x p32, K = c16)
            long nw = (long)o32 * c16;
            pad_w_kernel<<<(int)((nw + 255) / 256), 256, 0, stream>>>(
                W[s], Wp, O, C, ldW[s], 0, o32, c16);
            long waves = (long)Bn * (o32 >> 5) * (p32 >> 5);
            gemm_wmma_f32<false><<<(int)((waves + 7) / 8), 256, 0, stream>>>(
                Wp, Xp[s], Zd, Bn, o32, p32, c16, c16, p32, p32,
                0L, (long)d16 * p32, (long)o32 * p32);
            if (s == 1) {   // Zc = W[:, C:2C] @ X
                pad_w_kernel<<<(int)((nw + 255) / 256), 256, 0, stream>>>(
                    W[s], Wcp, O, C, ldW[s], C, o32, c16);
                gemm_wmma_f32<false><<<(int)((waves + 7) / 8), 256, 0, stream>>>(
                    Wcp, Xp[s], Zc, Bn, o32, p32, c16, c16, p32, p32,
                    0L, (long)d16 * p32, (long)o32 * p32);
            }
        }
        {   // S = (stage2 ? Zc : 0) - Zd over padded extent
            long n = (long)Bn * o32 * p32;
            s_kernel<<<(int)((n + 255) / 256), 256, 0, stream>>>(Zd, Zc, S, n,
                                                                 s == 1 ? 1 : 0);
        }
        bn_stats_graph<<<O, 256, 0, stream>>>(Zd, S, idx, mean, istd,
                                              Bn, O, P, kNN, p32, (long)o32 * p32);
        {   // fused BN + leaky + max_k, writing next stage's padded X (or x4)
            long osB; int osO, osP;
            if (s < 3) { osB = (long)D16[s + 1] * P32[s + 1]; osO = 1; osP = P32[s + 1]; }
            else       { osB = (long)O * P;                   osO = P; osP = 1; }
            long n = (long)Bn * O * P;
            apply_graph<<<(int)((n + 255) / 256), 256, 0, stream>>>(
                Zd, S, idx, mean, istd, Gm[s], Bt[s], Xnext[s],
                Bn, O, P, kNN, p32, (long)o32 * p32, osB, osO, osP);
        }
    }

    {   // head: pad W5 to 416x400, then 416x64x400 WMMA GEMM
        long n = (long)Bn * 400 * 64;
        headz_kernel<<<(int)((n + 255) / 256), 256, 0, stream>>>(X3p, X4, Zh, Bn);
        long nw = 416L * 400;
        pad_w_kernel<<<(int)((nw + 255) / 256), 256, 0, stream>>>(
            W5, W5p, 400, 400, 400, 0, 416, 400);
        long waves = (long)Bn * (416 / 32) * (64 / 32);
        gemm_wmma_f32<false><<<(int)((waves + 7) / 8), 256, 0, stream>>>(
            W5p, Zh, Yh, Bn, 416, 64, 400, 400, 64, 64,
            0L, 400L * 64, 416L * 64);
        bn_stats_head<<<400, 256, 0, stream>>>(Yh, mean, istd, Bn, 400, 64, 416L * 64);
        int tot = Bn * 400;
        head_out_kernel<<<(tot + 255) / 256, 256, 0, stream>>>(
            Yh, mean, istd, g5, b5, (float*)d_out, Bn, 400, 64, 416L * 64);
    }
}